// CorrelationLayer_56298431316460
// MI455X (gfx1250) — compile-verified
//
#include <hip/hip_runtime.h>

typedef float v2f __attribute__((ext_vector_type(2)));
typedef float v8f __attribute__((ext_vector_type(8)));
typedef int   v4i __attribute__((ext_vector_type(4)));
typedef int   v8i __attribute__((ext_vector_type(8)));
typedef unsigned int       u32;
typedef unsigned long long u64;

constexpr int B_   = 8;
constexpr int C_   = 256;
constexpr int H_   = 128;
constexpr int W_   = 128;
constexpr int PAD_ = 4;
constexpr int D_   = 9;          // 2*PAD+1 displacements per axis
constexpr int TW   = 16;         // pixels per tile (M dimension)
constexpr int WIN  = 24;         // window width = TW + 2*PAD
constexpr int K4_  = C_ / 4;     // 64 WMMA K-steps
constexpr float INV_C = 1.0f / 256.0f;

// ---------------------------------------------------------------------------
// TDM descriptor (D#) for one B window: 2D tile, 256 rows x 24 f32,
// row stride H*W elements, contiguous into LDS (= sB[c][j] layout).
// Group0: count=1 | lds_addr | global_addr(57b) | type=2.
// Group1: data_size=4B | tensor_dim0=24 | tensor_dim1=256 | tile_dim0=24 |
//         tile_dim1=256 | tensor_dim0_stride=16384.  (tensor==tile -> no OOB)
// ---------------------------------------------------------------------------
__device__ __forceinline__ void tdm_load_window(const float* tile, u32 ldsAddr)
{
    const u64 ga = (u64)(uintptr_t)tile;
    v4i g0;
    g0[0] = 1;                                                   // count=1, user desc
    g0[1] = (int)ldsAddr;                                        // lds_addr (bytes)
    g0[2] = (int)(u32)ga;                                        // global_addr[31:0]
    g0[3] = (int)(((u32)(ga >> 32) & 0x01FFFFFFu) | (2u << 30)); // addr[56:32]|type=2
    v8i g1;
    g1[0] = (int)(2u << 16);                                     // data_size=4B
    g1[1] = (int)(((u32)WIN) << 16);                             // tensor_dim0 lo16
    g1[2] = (int)(((u32)C_) << 16);                              // dim0 hi | tensor_dim1 lo16
    g1[3] = (int)(((u32)WIN) << 16);                             // dim1 hi | tile_dim0
    g1[4] = (int)((u32)C_);                                      // tile_dim1 | tile_dim2=0
    g1[5] = (int)((u32)(H_ * W_));                               // tensor_dim0_stride lo32
    g1[6] = 0;                                                   // stride0 hi | stride1 lo
    g1[7] = 0;                                                   // stride1 hi
    asm volatile("tensor_load_to_lds %0, %1" :: "s"(g0), "s"(g1) : "memory");
}

// Stage in1[b,:,h,w0:w0+16] into sA (coalesced float4), 64 threads.
__device__ __forceinline__ void stage_A(const float* __restrict__ in1,
                                        float (*sA)[TW],
                                        int b, int h, int w0, int tid)
{
    const float* src = in1 + (((size_t)b * C_) * H_ + h) * W_ + w0;
    #pragma unroll
    for (int i = 0; i < 16; ++i) {              // 1024 float4 / 64 threads
        int idx4 = i * 64 + tid;
        int c = idx4 >> 2;
        int g = idx4 & 3;
        const float4 v = *(const float4*)(src + (size_t)c * (H_ * W_) + 4 * g);
        *(float4*)&sA[c][4 * g] = v;
    }
}

// Band element (M, j) -> sOut[dx = j-M][M], scaled by 1/C (LDS transpose).
__device__ __forceinline__ void scatter_to_lds(float (*sOut)[TW], const v8f& acc,
                                               int wv, int half, int j)
{
    #pragma unroll
    for (int i = 0; i < 8; ++i) {
        int M  = i + 8 * half;                  // D layout: M = i + 8*half
        int dx = j - M;
        bool mine = (dx >= 0) && (dx <= 2 * PAD_) &&
                    (wv == 0 ? (j <= 15) : (j >= 16));
        if (mine) sOut[dx][M] = acc[i] * INV_C;
    }
}

// 36 threads emit 9 coalesced 64B row segments: out[b, dy*9+dx, h, w0..w0+15].
__device__ __forceinline__ void store_tile(float* __restrict__ out,
                                           const float (*sOut)[TW],
                                           int b, int dy, int h, int w0, int tid)
{
    if (tid < D_ * 4) {
        int dx = tid >> 2;
        int g  = tid & 3;
        float4 v = *(const float4*)&sOut[dx][4 * g];
        float* dst = out + (((size_t)b * (D_ * D_) + (size_t)(dy * D_ + dx)) * H_
                            + h) * W_ + w0 + 4 * g;
        *(float4*)dst = v;
    }
}

// ---------------------------------------------------------------------------
// Interior kernel: wb in 1..6 -- full window in-bounds in w.
// TDM double-buffered B windows; A fragments resident in 128 VGPRs.
// ---------------------------------------------------------------------------
__global__ __launch_bounds__(64)
void corr_wmma_interior(const float* __restrict__ in1,
                        const float* __restrict__ in2,
                        float* __restrict__ out)
{
    __shared__ float sA[C_][TW];       // 16 KB
    __shared__ float sB[2][C_][WIN];   // 48 KB double-buffered TDM target
    __shared__ float sOut[D_][TW];     // 576 B output transpose tile

    const int tid = threadIdx.x;
    const int wb  = (blockIdx.x % 6) + 1;          // 1..6
    const int h   = (blockIdx.x / 6) & 127;
    const int b   = blockIdx.x / (6 * 128);
    const int w0  = wb * TW;

    stage_A(in1, sA, b, h, w0, tid);

    const int lane = tid & 31;
    const int wv   = tid >> 5;
    const int half = lane >> 4;
    const int nn   = lane & 15;
    const int j    = wv * 8 + nn;

    const float* in2b = in2 + ((size_t)b * C_) * (H_ * W_);

    // Kick off dy=0 window; it streams while we fill the A register file.
    if (wv == 0 && (h - PAD_) >= 0) {
        tdm_load_window(in2b + (size_t)(h - PAD_) * W_ + (w0 - PAD_),
                        (u32)(uintptr_t)&sB[0][0][0]);
    }

    __syncthreads();                   // sA visible

    // A fragments resident across all 9 dy iterations (64 x v2f = 128 VGPRs).
    v2f aReg[K4_];
    #pragma unroll
    for (int k4 = 0; k4 < K4_; ++k4) {
        aReg[k4][0] = sA[4 * k4 + 2 * half + 0][nn];
        aReg[k4][1] = sA[4 * k4 + 2 * half + 1][nn];
    }

    for (int dy = 0; dy < D_; ++dy) {
        const int  buf = dy & 1;
        const int  h2  = h + dy - PAD_;
        const bool ok  = (h2 >= 0) && (h2 < H_);

        if (wv == 0) __builtin_amdgcn_s_wait_tensorcnt(0);  // window(dy) landed
        __syncthreads();            // TDM data visible; prior sOut readers done
        const int h3 = h2 + 1;
        if (wv == 0 && dy + 1 < D_ && h3 >= 0 && h3 < H_) { // overlap dy+1 DMA
            tdm_load_window(in2b + (size_t)h3 * W_ + (w0 - PAD_),
                            (u32)(uintptr_t)&sB[buf ^ 1][0][0]);
        }

        v8f acc = {};
        if (ok) {                       // uniform: EXEC all-ones through WMMA
            const float (*bw)[WIN] = sB[buf];
            #pragma unroll
            for (int k4 = 0; k4 < K4_; ++k4) {      // 64 x v_wmma, 1 ds_load each
                v2f bb;
                bb[0] = bw[4 * k4 + half + 0][j];
                bb[1] = bw[4 * k4 + half + 2][j];
                acc = __builtin_amdgcn_wmma_f32_16x16x4_f32(
                          false, aReg[k4], false, bb, (short)0, acc, false, false);
            }
        }

        scatter_to_lds(sOut, acc, wv, half, j);
        __syncthreads();
        store_tile(out, sOut, b, dy, h, w0, tid);
    }
}

// ---------------------------------------------------------------------------
// Edge kernel: wb in {0,7} -- needs zero padding in w; manual staged loads.
// ---------------------------------------------------------------------------
__global__ __launch_bounds__(64)
void corr_wmma_edge(const float* __restrict__ in1,
                    const float* __restrict__ in2,
                    float* __restrict__ out)
{
    __shared__ float sA[C_][TW];       // 16 KB
    __shared__ float sB[C_][WIN];      // 24 KB single buffer
    __shared__ float sOut[D_][TW];     // 576 B output transpose tile

    const int tid = threadIdx.x;
    const int wb  = (blockIdx.x & 1) * 7;          // 0 or 7
    const int h   = (blockIdx.x >> 1) & 127;
    const int b   = blockIdx.x >> 8;
    const int w0  = wb * TW;

    stage_A(in1, sA, b, h, w0, tid);

    const int lane = tid & 31;
    const int wv   = tid >> 5;
    const int half = lane >> 4;
    const int nn   = lane & 15;
    const int j    = wv * 8 + nn;

    const float* in2b = in2 + ((size_t)b * C_) * (H_ * W_);

    for (int dy = 0; dy < D_; ++dy) {
        const int  h2 = h + dy - PAD_;
        const bool ok = (h2 >= 0) && (h2 < H_);

        __syncthreads();                // prior sB/sOut readers done
        if (ok) {                       // bounds-checked zero-padded staging
            const float* srcB = in2b + (size_t)h2 * W_;
            for (int i = 0; i < 24; ++i) {          // 1536 float4-groups / 64 thr
                int idx4 = i * 64 + tid;
                int c = idx4 / 6;
                int g = idx4 % 6;
                int w = w0 - PAD_ + 4 * g;
                float4 v = make_float4(0.f, 0.f, 0.f, 0.f);
                const float* p = srcB + (size_t)c * (H_ * W_);
                if (w >= 0 && w + 3 < W_) {
                    v = *(const float4*)(p + w);
                } else {
                    if (w + 0 >= 0 && w + 0 < W_) v.x = p[w + 0];
                    if (w + 1 >= 0 && w + 1 < W_) v.y = p[w + 1];
                    if (w + 2 >= 0 && w + 2 < W_) v.z = p[w + 2];
                    if (w + 3 >= 0 && w + 3 < W_) v.w = p[w + 3];
                }
                *(float4*)&sB[c][4 * g] = v;
            }
        }
        __syncthreads();

        v8f acc = {};
        if (ok) {
            #pragma unroll 16
            for (int kk = 0; kk < C_; kk += 4) {
                v2f a, bb;
                a[0]  = sA[kk + 2 * half + 0][nn];
                a[1]  = sA[kk + 2 * half + 1][nn];
                bb[0] = sB[kk + half + 0][j];
                bb[1] = sB[kk + half + 2][j];
                acc = __builtin_amdgcn_wmma_f32_16x16x4_f32(
                          false, a, false, bb, (short)0, acc, false, false);
            }
        }

        scatter_to_lds(sOut, acc, wv, half, j);
        __syncthreads();
        store_tile(out, sOut, b, dy, h, w0, tid);
    }
}

extern "C" void kernel_launch(void* const* d_in, const int* in_sizes, int n_in,
                              void* d_out, int out_size, void* d_ws, size_t ws_size,
                              hipStream_t stream) {
    const float* in1 = (const float*)d_in[0];
    const float* in2 = (const float*)d_in[1];
    float* outp = (float*)d_out;
    corr_wmma_interior<<<dim3(B_ * H_ * 6), dim3(64), 0, stream>>>(in1, in2, outp);
    corr_wmma_edge    <<<dim3(B_ * H_ * 2), dim3(64), 0, stream>>>(in1, in2, outp);
}